// LightGCNLayer_7129645711539
// MI455X (gfx1250) — compile-verified
//
#include <hip/hip_runtime.h>
#include <hip/hip_bf16.h>
#include <stdint.h>

#define TPB 256
#define D_FEAT 128
#define F4_PER_ROW (D_FEAT / 4)   // 32 float4 per feature row: one per lane of a wave32
#define WAVES_PER_BLOCK (TPB / 32)
#define GB 4                      // neighbors per async batch (double-buffered)

// 16-byte int vector type expected by the async-to-LDS builtin.
typedef int v4i __attribute__((vector_size(16)));

// ---------------------------------------------------------------------------
// CDNA5 async-copy primitives (gfx1250): GLOBAL_LOAD_ASYNC_TO_LDS_B128 moves
// 16 B/lane (512 B/wave) from global memory straight into LDS, tracked with
// the per-wave ASYNCcnt (completes in order). Builtin if available, else
// inline asm per CDNA5 ISA.
// ---------------------------------------------------------------------------
__device__ __forceinline__ void async_gather_b128(const float4* g, float4* l) {
#if __has_builtin(__builtin_amdgcn_global_load_async_to_lds_b128)
  __builtin_amdgcn_global_load_async_to_lds_b128(
      (__attribute__((address_space(1))) v4i*)(void*)g,
      (__attribute__((address_space(3))) v4i*)(void*)l, 0, 0);
#else
  asm volatile("global_load_async_to_lds_b128 %0, %1, off"
               :: "v"((__attribute__((address_space(3))) v4i*)(void*)l), "v"(g)
               : "memory");
#endif
}

template <int N>
__device__ __forceinline__ void wait_asynccnt() {
#if __has_builtin(__builtin_amdgcn_s_wait_asynccnt)
  __builtin_amdgcn_s_wait_asynccnt(N);
#else
  asm volatile("s_wait_asynccnt %0" :: "i"(N) : "memory");
#endif
}

// n is wave-uniform; ASYNCcnt waits need immediates, so switch on small n.
__device__ __forceinline__ void wait_asynccnt_dyn(int n) {
  switch (n) {
    case 0:  wait_asynccnt<0>(); break;
    case 1:  wait_asynccnt<1>(); break;
    case 2:  wait_asynccnt<2>(); break;
    case 3:  wait_asynccnt<3>(); break;
    default: wait_asynccnt<GB>(); break;
  }
}

// ---------------------------------------------------------------------------
// Gather + mean (hot kernel, first in file so the disasm snippet shows it).
// One wave32 per output node; lane L owns features [4L, 4L+4).
// Double-buffered async pipeline: while batch i is being accumulated from
// LDS, batch i+1 (GB neighbor rows) is in flight via ASYNCcnt.
// Writes every output element (zero-degree nodes write 0 -> no pre-zeroing).
// ---------------------------------------------------------------------------
__global__ void gather_mean_kernel(const float4* __restrict__ feat,
                                   const int* __restrict__ off,
                                   const int* __restrict__ adj,
                                   float4* __restrict__ out, int n_nodes) {
  __shared__ float4 smem[WAVES_PER_BLOCK][2][GB][32];   // 32 KB / block

  int gid = blockIdx.x * blockDim.x + threadIdx.x;
  int node = gid >> 5;
  int wv = threadIdx.x >> 5;
  int lane = threadIdx.x & 31;
  if (node >= n_nodes) return;

  int beg = off[node];
  int deg = off[node + 1] - beg;    // wave-uniform

  float4 acc = {0.f, 0.f, 0.f, 0.f};

  // Prologue: issue batch 0.
  int c0 = deg < GB ? deg : GB;
#pragma unroll
  for (int j = 0; j < GB; ++j) {
    if (j < c0) {
      int nb = adj[beg + j];
      async_gather_b128(feat + (size_t)nb * F4_PER_ROW + lane, &smem[wv][0][j][lane]);
    }
  }

  int k = 0, cur = 0;
  while (k < deg) {
    int rem = deg - k;
    int cthis = rem < GB ? rem : GB;
    int rem2 = rem - cthis;
    int cnext = rem2 < GB ? rem2 : GB;

    // Issue next batch into the other buffer (stays in flight while we work).
#pragma unroll
    for (int j = 0; j < GB; ++j) {
      if (j < cnext) {
        int nb = adj[beg + k + cthis + j];
        async_gather_b128(feat + (size_t)nb * F4_PER_ROW + lane,
                          &smem[wv][cur ^ 1][j][lane]);
      }
    }

    // Async loads retire in order: ASYNCcnt <= cnext  =>  batch `cur` landed.
    wait_asynccnt_dyn(cnext);

    // Accumulate batch `cur` from LDS (ds_load_b128 per neighbor per lane).
#pragma unroll
    for (int j = 0; j < GB; ++j) {
      if (j < cthis) {
        float4 v = smem[wv][cur][j][lane];
        acc.x += v.x; acc.y += v.y; acc.z += v.z; acc.w += v.w;
      }
    }

    k += cthis;
    cur ^= 1;
  }

  float inv = (deg > 0) ? (1.0f / (float)deg) : 0.0f;   // mean; 0-degree -> 0
  float4 r;
  r.x = acc.x * inv; r.y = acc.y * inv; r.z = acc.z * inv; r.w = acc.w * inv;
  out[(size_t)node * F4_PER_ROW + lane] = r;
}

// ---------------------------------------------------------------------------
// Utility: zero arrays
// ---------------------------------------------------------------------------
__global__ void zero_i32_kernel(int* __restrict__ p, int n) {
  int i = blockIdx.x * blockDim.x + threadIdx.x;
  if (i < n) p[i] = 0;
}

__global__ void zero_f32_kernel(float* __restrict__ p, size_t n) {
  size_t i = (size_t)blockIdx.x * blockDim.x + threadIdx.x;
  if (i < n) p[i] = 0.0f;
}

// ---------------------------------------------------------------------------
// Pass 1: per-node degree histogram (int atomics, counters are L2-resident)
// ---------------------------------------------------------------------------
__global__ void hist_kernel(const int* __restrict__ src, const int* __restrict__ dst,
                            int* __restrict__ cnt_user, int* __restrict__ cnt_item, int E) {
  int e = blockIdx.x * blockDim.x + threadIdx.x;
  if (e < E) {
    atomicAdd(&cnt_user[src[e]], 1);
    atomicAdd(&cnt_item[dst[e]], 1);
  }
}

// ---------------------------------------------------------------------------
// Pass 2: exclusive prefix sum, single 1024-thread workgroup, chunked
// Hillis-Steele in LDS. n <= ~100K -> ~100 chunks, negligible runtime.
// off has n+1 entries; off[n] = total (== E).
// ---------------------------------------------------------------------------
__global__ void __launch_bounds__(1024) scan_kernel(const int* __restrict__ cnt,
                                                    int* __restrict__ off, int n) {
  __shared__ int buf[1024];
  __shared__ int s_carry;
  const int tid = threadIdx.x;
  if (tid == 0) s_carry = 0;
  __syncthreads();
  for (int base = 0; base < n; base += 1024) {
    int i = base + tid;
    int v = (i < n) ? cnt[i] : 0;
    int c = s_carry;            // carry from previous chunk (barrier-protected)
    buf[tid] = v;
    __syncthreads();
#pragma unroll
    for (int s = 1; s < 1024; s <<= 1) {
      int t = (tid >= s) ? buf[tid - s] : 0;
      __syncthreads();
      buf[tid] += t;
      __syncthreads();
    }
    int incl = buf[tid];
    if (i < n) off[i] = c + (incl - v);   // exclusive
    __syncthreads();
    if (tid == 1023) s_carry = c + incl;  // chunk total
    __syncthreads();
  }
  if (tid == 0) off[n] = s_carry;
}

// ---------------------------------------------------------------------------
// Pass 3: fill adjacency lists. adj_user[pos] = neighbor ITEM ids grouped by
// user; adj_item[pos] = neighbor USER ids grouped by item.
// ---------------------------------------------------------------------------
__global__ void fill_kernel(const int* __restrict__ src, const int* __restrict__ dst,
                            const int* __restrict__ off_user, const int* __restrict__ off_item,
                            int* __restrict__ cur_user, int* __restrict__ cur_item,
                            int* __restrict__ adj_user, int* __restrict__ adj_item, int E) {
  int e = blockIdx.x * blockDim.x + threadIdx.x;
  if (e < E) {
    int u = src[e], it = dst[e];
    int p = off_user[u] + atomicAdd(&cur_user[u], 1);
    adj_user[p] = it;
    int q = off_item[it] + atomicAdd(&cur_item[it], 1);
    adj_item[q] = u;
  }
}

// ---------------------------------------------------------------------------
// Fallback path (used only if ws_size cannot hold the CSR build, ~13.4 MB):
// wave-per-edge atomic scatter + normalize. Needs only (n_users+n_items)
// floats of workspace for degree counters.
// ---------------------------------------------------------------------------
__global__ void scatter_kernel(const float4* __restrict__ user_feat,
                               const float4* __restrict__ item_feat,
                               const int* __restrict__ src, const int* __restrict__ dst,
                               float* __restrict__ out_user, float* __restrict__ out_item,
                               float* __restrict__ cnt_user, float* __restrict__ cnt_item, int E) {
  int gid = blockIdx.x * blockDim.x + threadIdx.x;
  int e = gid >> 5;
  int lane = threadIdx.x & 31;
  if (e >= E) return;
  int u = src[e], it = dst[e];
  float4 uf = user_feat[(size_t)u * F4_PER_ROW + lane];
  float4 vf = item_feat[(size_t)it * F4_PER_ROW + lane];
  float* oi = out_item + (size_t)it * D_FEAT + lane * 4;
  atomicAdd(oi + 0, uf.x); atomicAdd(oi + 1, uf.y);
  atomicAdd(oi + 2, uf.z); atomicAdd(oi + 3, uf.w);
  float* ou = out_user + (size_t)u * D_FEAT + lane * 4;
  atomicAdd(ou + 0, vf.x); atomicAdd(ou + 1, vf.y);
  atomicAdd(ou + 2, vf.z); atomicAdd(ou + 3, vf.w);
  if (lane == 0) {
    atomicAdd(&cnt_item[it], 1.0f);
    atomicAdd(&cnt_user[u], 1.0f);
  }
}

__global__ void normalize_kernel(float* __restrict__ out, const float* __restrict__ cnt,
                                 int n_nodes) {
  size_t i = (size_t)blockIdx.x * blockDim.x + threadIdx.x;
  size_t total = (size_t)n_nodes * D_FEAT;
  if (i < total) {
    float c = cnt[i >> 7];          // i / 128
    out[i] = (c > 0.f) ? out[i] / c : 0.0f;
  }
}

// ---------------------------------------------------------------------------
// Host-side launcher (graph-capture safe: no mallocs, no syncs, all on stream)
// ---------------------------------------------------------------------------
extern "C" void kernel_launch(void* const* d_in, const int* in_sizes, int n_in,
                              void* d_out, int out_size, void* d_ws, size_t ws_size,
                              hipStream_t stream) {
  const float* user_feat = (const float*)d_in[0];
  const float* item_feat = (const float*)d_in[1];
  const int*   edge_src  = (const int*)d_in[2];
  const int*   edge_dst  = (const int*)d_in[3];

  const int n_users = in_sizes[0] / D_FEAT;
  const int n_items = in_sizes[1] / D_FEAT;
  const int E       = in_sizes[2];

  float* out_user = (float*)d_out;                         // [n_users, 128]
  float* out_item = out_user + (size_t)n_users * D_FEAT;   // [n_items, 128]

  const size_t need_csr = ((size_t)(n_users + n_items)              // cnt/cursor
                         + (size_t)(n_users + 1) + (n_items + 1)    // offsets
                         + (size_t)E * 2) * sizeof(int);            // adjacency

  if (ws_size >= need_csr) {
    // ---- CSR-gather path (preferred: no f32 atomics on the hot loop) ----
    int* cnt_user = (int*)d_ws;                 // n_users (reused as cursor)
    int* cnt_item = cnt_user + n_users;         // n_items (reused as cursor)
    int* off_user = cnt_item + n_items;         // n_users + 1
    int* off_item = off_user + (n_users + 1);   // n_items + 1
    int* adj_user = off_item + (n_items + 1);   // E  (item ids grouped by user)
    int* adj_item = adj_user + E;               // E  (user ids grouped by item)

    const int ncnt = n_users + n_items;
    zero_i32_kernel<<<(ncnt + TPB - 1) / TPB, TPB, 0, stream>>>(cnt_user, ncnt);
    hist_kernel<<<(E + TPB - 1) / TPB, TPB, 0, stream>>>(edge_src, edge_dst,
                                                         cnt_user, cnt_item, E);
    scan_kernel<<<1, 1024, 0, stream>>>(cnt_user, off_user, n_users);
    scan_kernel<<<1, 1024, 0, stream>>>(cnt_item, off_item, n_items);
    zero_i32_kernel<<<(ncnt + TPB - 1) / TPB, TPB, 0, stream>>>(cnt_user, ncnt);
    fill_kernel<<<(E + TPB - 1) / TPB, TPB, 0, stream>>>(edge_src, edge_dst,
                                                         off_user, off_item,
                                                         cnt_user, cnt_item,
                                                         adj_user, adj_item, E);

    gather_mean_kernel<<<(n_users + WAVES_PER_BLOCK - 1) / WAVES_PER_BLOCK, TPB, 0, stream>>>(
        (const float4*)item_feat, off_user, adj_user, (float4*)out_user, n_users);
    gather_mean_kernel<<<(n_items + WAVES_PER_BLOCK - 1) / WAVES_PER_BLOCK, TPB, 0, stream>>>(
        (const float4*)user_feat, off_item, adj_item, (float4*)out_item, n_items);
  } else {
    // ---- Fallback: atomic scatter (only degree counters in ws) ----
    float* cnt_user_f = (float*)d_ws;            // n_users
    float* cnt_item_f = cnt_user_f + n_users;    // n_items

    size_t out_total = (size_t)(n_users + n_items) * D_FEAT;
    zero_f32_kernel<<<(unsigned)((out_total + TPB - 1) / TPB), TPB, 0, stream>>>(
        (float*)d_out, out_total);
    zero_f32_kernel<<<(n_users + n_items + TPB - 1) / TPB, TPB, 0, stream>>>(
        cnt_user_f, (size_t)(n_users + n_items));

    size_t scatter_threads = (size_t)E * 32;
    scatter_kernel<<<(unsigned)((scatter_threads + TPB - 1) / TPB), TPB, 0, stream>>>(
        (const float4*)user_feat, (const float4*)item_feat, edge_src, edge_dst,
        out_user, out_item, cnt_user_f, cnt_item_f, E);

    size_t nu_total = (size_t)n_users * D_FEAT;
    size_t ni_total = (size_t)n_items * D_FEAT;
    normalize_kernel<<<(unsigned)((nu_total + TPB - 1) / TPB), TPB, 0, stream>>>(
        out_user, cnt_user_f, n_users);
    normalize_kernel<<<(unsigned)((ni_total + TPB - 1) / TPB), TPB, 0, stream>>>(
        out_item, cnt_item_f, n_items);
  }
}